// SpatialTransformer_65506841199045
// MI455X (gfx1250) — compile-verified
//
#include <hip/hip_runtime.h>

// Spatial transformer (bilinear grid sample), B=32, H=W=256, C=32, f32 NHWC.
// Memory-bound (~540 MB HBM traffic -> ~23 us @ 23.3 TB/s). Strategy:
//  - coordinate einsum theta(2x3) x grid(3x16) done with V_WMMA_F32_16X16X4_F32
//  - 8 lanes per pixel, one float4 channel-group per lane -> all gather taps are
//    contiguous 128B (global_load_b128), output stores 512B/wave contiguous
//  - non-temporal b128 stores keep the 268MB write stream out of L2 so the
//    per-batch 8.4MB image stays resident for the 4-tap gather reuse.

typedef __attribute__((ext_vector_type(2))) float v2f;
typedef __attribute__((ext_vector_type(4))) float v4f;
typedef __attribute__((ext_vector_type(8))) float v8f;

#define ST_B 32
#define ST_H 256
#define ST_W 256
#define ST_C 32
// pixels per batch = 65536, 16-pixel groups per batch = 4096

__global__ __launch_bounds__(256) void st_wmma_kernel(
    const float* __restrict__ theta,
    const float* __restrict__ imgs,
    float* __restrict__ out)
{
    const int lane = threadIdx.x & 31;
    const int wave = threadIdx.x >> 5;
    const int g    = blockIdx.x * 8 + wave;                      // global 16-pixel group
    const int b    = __builtin_amdgcn_readfirstlane(g >> 12);    // 4096 groups / batch
    const int pbase = (g & 4095) << 4;                           // first in-batch pixel

    // ---- theta for this batch (wave-uniform -> scalar loads) ----
    const float* th = theta + b * 6;
    const float t0 = th[0], t1 = th[1], t2 = th[2];
    const float t3 = th[3], t4 = th[4], t5 = th[5];

    // ---- A matrix 16x4 f32:
    //   row M=0 : [t0 t1 t2 0]  (x transform)
    //   row M=1 : [t3 t4 t5 0]  (y transform)
    // layout: lanes 0-15 -> M=lane, (K0,K1) in (v.x,v.y); lanes 16-31 -> M=lane-16, (K2,K3)
    v2f a; a.x = 0.0f; a.y = 0.0f;
    if (lane == 0)  { a.x = t0; a.y = t1; }
    if (lane == 1)  { a.x = t3; a.y = t4; }
    if (lane == 16) { a.x = t2; }
    if (lane == 17) { a.x = t5; }

    // ---- B matrix 4x16 f32: column n = [x_t(p), y_t(p), 1, 0], p = pbase + n
    // layout: lanes 0-15 -> N=lane, (K0,K1); lanes 16-31 -> N=lane-16, (K2,K3)
    v2f bm;
    {
        const int   n  = lane & 15;
        const int   p  = pbase + n;
        const float xt = fmaf((float)(p & (ST_W - 1)), 2.0f / (ST_W - 1), -1.0f);
        const float yt = fmaf((float)(p >> 8),         2.0f / (ST_H - 1), -1.0f);
        if (lane < 16) { bm.x = xt;   bm.y = yt;   }
        else           { bm.x = 1.0f; bm.y = 0.0f; }
    }

    v8f c = {0.f, 0.f, 0.f, 0.f, 0.f, 0.f, 0.f, 0.f};
    // D = A x B : D row 0 (lanes 0-15) = sampled x for 16 pixels, row 1 = sampled y
    v8f d = __builtin_amdgcn_wmma_f32_16x16x4_f32(
        /*neg_a=*/false, a, /*neg_b=*/false, bm,
        /*c_mod=*/(short)0, c, /*reuse_a=*/false, /*reuse_b=*/false);
    const float xsAll = d[0];
    const float ysAll = d[1];

    // ---- gather phase: 8 lanes per pixel (one float4 channel group each) ----
    const int cg = lane & 7;
    const v4f* __restrict__ img_b =
        (const v4f*)imgs + (size_t)b * (ST_H * ST_W * (ST_C / 4));
    v4f* __restrict__ out_b =
        (v4f*)out + (size_t)b * (ST_H * ST_W * (ST_C / 4)) + (size_t)pbase * 8 + cg;

#pragma unroll
    for (int it = 0; it < 4; ++it) {
        const int   pg = it * 4 + (lane >> 3);        // pixel 0..15 within group
        const float xs = __shfl(xsAll, pg, 32);
        const float ys = __shfl(ysAll, pg, 32);

        const float x = fmaf(xs, 128.0f, 128.0f);     // (xs + 1) * W * 0.5
        const float y = fmaf(ys, 128.0f, 128.0f);     // (ys + 1) * H * 0.5
        float x0 = floorf(x), y0 = floorf(y);
        float x1 = x0 + 1.0f, y1 = y0 + 1.0f;
        x0 = fminf(fmaxf(x0, 0.0f), 255.0f);
        x1 = fminf(fmaxf(x1, 0.0f), 255.0f);
        y0 = fminf(fmaxf(y0, 0.0f), 255.0f);
        y1 = fminf(fmaxf(y1, 0.0f), 255.0f);
        const float wa = (y1 - y) * (x1 - x);
        const float wb = (y1 - y) * (x - x0);
        const float wc = (y - y0) * (x1 - x);
        const float wd = (y - y0) * (x - x0);
        const int x0i = (int)x0, x1i = (int)x1;
        const int y0i = (int)y0, y1i = (int)y1;

        const v4f va = img_b[(y0i * ST_W + x0i) * 8 + cg];   // global_load_b128 x4
        const v4f vb = img_b[(y0i * ST_W + x1i) * 8 + cg];
        const v4f vc = img_b[(y1i * ST_W + x0i) * 8 + cg];
        const v4f vd = img_b[(y1i * ST_W + x1i) * 8 + cg];

        v4f acc = va * wa;
        acc += vb * wb;
        acc += vc * wc;
        acc += vd * wd;

        __builtin_nontemporal_store(acc, out_b + pg * 8);    // NT global_store_b128
    }
}

extern "C" void kernel_launch(void* const* d_in, const int* in_sizes, int n_in,
                              void* d_out, int out_size, void* d_ws, size_t ws_size,
                              hipStream_t stream) {
    const float* theta = (const float*)d_in[0];
    const float* imgs  = (const float*)d_in[1];
    float*       out   = (float*)d_out;

    // B*H*W = 2,097,152 pixels -> 131,072 groups of 16 -> 8 waves/block -> 16384 blocks
    const int groups = ST_B * ST_H * ST_W / 16;
    dim3 grid(groups / 8), block(256);
    hipLaunchKernelGGL(st_wmma_kernel, grid, block, 0, stream, theta, imgs, out);
}